// CROSS_DOMAIN_FSAR_30683246362893
// MI455X (gfx1250) — compile-verified
//
#include <hip/hip_runtime.h>
#include <stdint.h>

// OTAM soft-DTW cumulative distance for MI455X (gfx1250).
// B = 256*64 = 16384 independent 48x50 DPs, one per lane.
// One wave per workgroup; TDM streams each DP row (strided 2D tile) into
// double-buffered LDS; s_wait_tensorcnt orders DMA vs compute (no barriers
// needed for single-wave workgroups).

typedef __attribute__((ext_vector_type(4))) unsigned int u32x4;
typedef __attribute__((ext_vector_type(4))) int          i32x4;
typedef __attribute__((ext_vector_type(8))) int          i32x8;

#define NQ    256
#define NS    64
#define LROWS 48
#define MCOLS 48
#define M2    50          // padded row length (zeros at both ends)
#define NP    32          // problems per block == one wave32

// exp(-2*d) = exp2(-KEXP*d), KEXP = 2*log2(e)
#define KEXP 2.8853900817779268f
// 0.5 * ln(s) = KLOG * log2(s), KLOG = 0.5*ln(2)
#define KLOG 0.34657359027997264f

// Issue one TDM load: 2D tile of NP lines x MCOLS floats, line stride = one
// problem block (LROWS*MCOLS floats), destination LDS at lds_off.
__device__ __forceinline__ void tdm_load_rows(uint32_t lds_off, uint64_t gaddr) {
  u32x4 g0;
  g0.x = 1u;                                  // count=1, user mode, no gather
  g0.y = lds_off;                             // LDS byte address
  g0.z = (uint32_t)gaddr;                     // global_addr[31:0]
  g0.w = (uint32_t)((gaddr >> 32) & 0x1FFFFFFull) | (2u << 30);  // addr[56:32] | type=2

  i32x8 g1;
  g1[0] = (int)(2u << 16);                    // data_size = 2 (4 bytes); mask=0
  g1[1] = (int)(((uint32_t)MCOLS) << 16);     // tensor_dim0[15:0] in bits[63:48]
  g1[2] = (int)(((uint32_t)NP) << 16);        // tensor_dim0 hi=0 | tensor_dim1[15:0]
  g1[3] = (int)(((uint32_t)MCOLS) << 16);     // tensor_dim1 hi=0 | tile_dim0
  g1[4] = (int)NP;                            // tile_dim1=NP | tile_dim2=0
  g1[5] = (int)(LROWS * MCOLS);               // tensor_dim0_stride = 2304 elems
  g1[6] = 0;                                  // stride hi | dim1_stride lo
  g1[7] = 0;                                  // dim1_stride hi

  i32x4 g2 = {1, 0, 0, 0};                    // tensor_dim2=1 (unused), no iterate
  i32x4 g3 = {0, 0, 0, 0};

#if defined(__clang_major__) && (__clang_major__ >= 23)
  i32x8 gpad = {0, 0, 0, 0, 0, 0, 0, 0};
  __builtin_amdgcn_tensor_load_to_lds(g0, g1, g2, g3, gpad, 0);
#else
  __builtin_amdgcn_tensor_load_to_lds(g0, g1, g2, g3, 0);
#endif
}

__global__ __launch_bounds__(NP) void CROSS_DOMAIN_FSAR_otam_kernel(
    const float* __restrict__ dists, float* __restrict__ out) {
  __shared__ float buf[2][NP * MCOLS];

  const int t  = threadIdx.x;                 // lane = problem within block
  const int b0 = blockIdx.x * NP;             // first problem of this block

  // byte address of row 0 of this block's first problem
  const uint64_t base =
      (uint64_t)(uintptr_t)dists + (uint64_t)b0 * (LROWS * MCOLS) * 4ull;
  // LDS byte offsets of the two row buffers (low 32 bits of generic ptr)
  const uint32_t lds0 = (uint32_t)(uintptr_t)&buf[0][0];
  const uint32_t lds1 = (uint32_t)(uintptr_t)&buf[1][0];

  // Prologue: prefetch rows 0 and 1, then wait until row 0 has landed.
  tdm_load_rows(lds0, base);
  tdm_load_rows(lds1, base + (uint64_t)MCOLS * 4ull);
  __builtin_amdgcn_s_wait_tensorcnt(1);
  asm volatile("" ::: "memory");

  // Row 0: cum[0,m] = running sum of padded row (pad cols contribute 0).
  float prev[M2];
  {
    const float* r = &buf[0][t * MCOLS];
    float acc = 0.0f;
    prev[0] = 0.0f;
#pragma unroll
    for (int j = 0; j < MCOLS; ++j) {
      acc += r[j];
      prev[j + 1] = acc;
    }
    prev[M2 - 1] = acc;   // trailing pad adds 0
  }

  // DP rows 1..47 with double-buffered TDM prefetch of row l+1.
#pragma unroll 1
  for (int l = 1; l < LROWS; ++l) {
    if (l + 1 < LROWS) {
      // All DS reads of the buffer being overwritten were consumed last
      // iteration; drain DScnt before the DMA rewrites it, then keep at
      // most the new prefetch in flight.
      asm volatile("s_wait_dscnt 0x0" ::: "memory");
      const uint32_t dst = ((l + 1) & 1) ? lds1 : lds0;
      tdm_load_rows(dst, base + (uint64_t)(l + 1) * MCOLS * 4ull);
      __builtin_amdgcn_s_wait_tensorcnt(1);   // row l complete
    } else {
      __builtin_amdgcn_s_wait_tensorcnt(0);   // last row complete
    }
    asm volatile("" ::: "memory");

    const float* r = (l & 1) ? &buf[1][t * MCOLS] : &buf[0][t * MCOLS];

    float cprev = 0.0f;       // cur[l, 0] = 0
    float left  = prev[0];    // old prev[m-1], starts at prev[0] (== 0)
#pragma unroll
    for (int m = 1; m < M2; ++m) {
      const float dm  = (m <= MCOLS) ? r[m - 1] : 0.0f;  // padded d
      const float pm1 = left;       // old prev[m-1]
      const float pm  = prev[m];    // old prev[m]
      const bool edge = (m == 1) || (m == M2 - 1);

      // stabilized softmin: mn - lambda*log(sum exp(-(x-mn)/lambda))
      float mn = fminf(pm1, cprev);
      float s;
      if (edge) {
        mn = fminf(mn, pm);
        s = __builtin_amdgcn_exp2f((mn - pm1) * KEXP) +
            __builtin_amdgcn_exp2f((mn - cprev) * KEXP) +
            __builtin_amdgcn_exp2f((mn - pm) * KEXP);
      } else {
        s = __builtin_amdgcn_exp2f((mn - pm1) * KEXP) +
            __builtin_amdgcn_exp2f((mn - cprev) * KEXP);
      }
      const float c = dm + mn - KLOG * __builtin_amdgcn_logf(s);

      left    = pm;   // preserve old prev[m] for next cell's prev[m-1]
      prev[m] = c;    // write cur[m] in place
      cprev   = c;
    }
  }

  out[b0 + t] = prev[M2 - 1];
}

extern "C" void kernel_launch(void* const* d_in, const int* in_sizes, int n_in,
                              void* d_out, int out_size, void* d_ws, size_t ws_size,
                              hipStream_t stream) {
  const float* dists = (const float*)d_in[0];
  float*       out   = (float*)d_out;

  const int B = NQ * NS;                      // 16384 problems
  dim3 grid(B / NP), block(NP);               // 512 blocks x 1 wave32
  CROSS_DOMAIN_FSAR_otam_kernel<<<grid, block, 0, stream>>>(dists, out);
}